// KANLinear_51539607844
// MI455X (gfx1250) — compile-verified
//
#include <hip/hip_runtime.h>
#include <cmath>

// KANLinear: N=1024, IN=256, OUT=256, G=5, K=3 (G+K = 8 spline coeffs)
#define N_ROWS   1024
#define IN_DIM   256
#define OUT_DIM  256
#define PACK     16     // floats per staged element: [lo-half 8 | hi-half 8]

typedef __attribute__((ext_vector_type(2))) float v2f;
typedef __attribute__((ext_vector_type(4))) float v4f;
typedef __attribute__((ext_vector_type(8))) float v8f;

// Packed element layout (16 floats), k0..k11 = 12-coef vector (3 x K4 chunks):
//   [0..7]  : k0 k1 k4 k5 k8 k9 0 0    (consumed by lanes 0-15,  K-pair base 0)
//   [8..15] : k2 k3 k6 k7 k10 k11 0 0  (consumed by lanes 16-31, K-pair base 2)

// ---------------------------------------------------------------------------
// Kernel 0: per-grid-row reciprocals of Cox-de-Boor denominators.
// r[j-1][t] = 1/(g[t+j]-g[t]);  note rightDenom(j,t) == leftDenom(j,t+1).
// ---------------------------------------------------------------------------
__global__ void kan_recips(const float* __restrict__ grid, float* __restrict__ R) {
  int i = blockIdx.x * blockDim.x + threadIdx.x;
  if (i >= IN_DIM) return;
  float g[12];
#pragma unroll
  for (int t = 0; t < 12; ++t) g[t] = grid[i * 12 + t];
  float* r = R + i * 32;
#pragma unroll
  for (int j = 1; j <= 3; ++j)
#pragma unroll
    for (int t = 0; t + j <= 11; ++t)
      r[(j - 1) * 11 + t] = 1.0f / (g[t + j] - g[t]);
}

// ---------------------------------------------------------------------------
// Kernel 1: per-(n,i) features: [silu(xq), bsp0..bsp7], packed per lane-half.
// F[(i*N_ROWS + n)*16 + .]  (i-major: a wave's per-i reads are contiguous)
// ---------------------------------------------------------------------------
__global__ void kan_features(const float* __restrict__ x,
                             const float* __restrict__ grid,
                             const float* __restrict__ R,
                             float* __restrict__ F) {
  int idx = blockIdx.x * blockDim.x + threadIdx.x;   // idx = i*N_ROWS + n
  if (idx >= N_ROWS * IN_DIM) return;
  int n = idx & (N_ROWS - 1);
  int i = idx >> 10;

  float xv = x[n * IN_DIM + i];

  // ste_quant(x, tp=7, fp=6): floor to 1/64, wrap mod 128 centered -> [-1, 1)
  int ii = (int)floorf(xv * 64.0f);
  int w7 = (ii + 64) & 127;
  float xq = (float)(w7 - 64) * (1.0f / 64.0f);

  float s = xq / (1.0f + expf(-xq));   // silu

  float g[12];
#pragma unroll
  for (int t = 0; t < 12; ++t) g[t] = grid[i * 12 + t];

  float bas[11];
#pragma unroll
  for (int t = 0; t < 11; ++t)
    bas[t] = (xq >= g[t] && xq < g[t + 1]) ? 1.0f : 0.0f;

  const float* rbase = R + i * 32;
#pragma unroll
  for (int j = 1; j <= 3; ++j) {
    const float* r = rbase + (j - 1) * 11;
#pragma unroll
    for (int t = 0; t + j < 11; ++t) {
      float left  = (xq - g[t]) * r[t];
      float right = (g[t + j + 1] - xq) * r[t + 1];
      bas[t] = left * bas[t] + right * bas[t + 1];
    }
  }

  // k0=s, k1..k8=bas0..bas7, k9..k11=0 ; packed per lane-half
  float* dst = F + (size_t)idx * PACK;
  v4f p0 = {s,      bas[0], bas[3], bas[4]};   // k0 k1 k4 k5
  v4f p1 = {bas[7], 0.0f,   0.0f,   0.0f};     // k8 k9 pad pad
  v4f p2 = {bas[1], bas[2], bas[5], bas[6]};   // k2 k3 k6 k7
  v4f p3 = {0.0f,   0.0f,   0.0f,   0.0f};     // k10 k11 pad pad
  *(v4f*)(dst + 0)  = p0;
  *(v4f*)(dst + 4)  = p1;
  *(v4f*)(dst + 8)  = p2;
  *(v4f*)(dst + 12) = p3;
}

// ---------------------------------------------------------------------------
// Kernel 2: per-(o,i) fused coefficients, pre-scaled by 64 (exact pow2) so the
// WMMA output is already act*64.  C[(i*OUT_DIM + o)*16 + .]
// ---------------------------------------------------------------------------
__global__ void kan_coefs(const float* __restrict__ bw,
                          const float* __restrict__ sw,
                          const float* __restrict__ sc,
                          const float* __restrict__ sel,
                          float* __restrict__ C) {
  int idx = blockIdx.x * blockDim.x + threadIdx.x;   // idx = i*OUT_DIM + o
  if (idx >= OUT_DIM * IN_DIM) return;
  int o = idx & (OUT_DIM - 1);
  int i = idx >> 8;
  int oi = o * IN_DIM + i;

  float se = sel[oi];
  float scale = sc[oi];
  float c0 = (bw[oi] * se) * 64.0f;
  const float* swp = sw + (size_t)oi * 8;

  float c[8];
#pragma unroll
  for (int k = 0; k < 8; ++k) c[k] = ((swp[k] * scale) * se) * 64.0f;

  float* dst = C + (size_t)idx * PACK;
  v4f p0 = {c0,   c[0], c[3], c[4]};
  v4f p1 = {c[7], 0.0f, 0.0f, 0.0f};
  v4f p2 = {c[1], c[2], c[5], c[6]};
  v4f p3 = {0.0f, 0.0f, 0.0f, 0.0f};
  *(v4f*)(dst + 0)  = p0;
  *(v4f*)(dst + 4)  = p1;
  *(v4f*)(dst + 8)  = p2;
  *(v4f*)(dst + 12) = p3;
}

// ---------------------------------------------------------------------------
// Kernel 3: one wave per 16x16 (n,o) tile; per i: b128+b64 loads for A and B,
// 3x V_WMMA_F32_16X16X4_F32, then floor and *float* accumulate (integer-valued
// floats, |sum| << 2^24 -> exact; single int convert + mod-2^16 wrap at end).
// ---------------------------------------------------------------------------
__global__ void __launch_bounds__(256) kan_wmma(const float* __restrict__ F,
                                                const float* __restrict__ C,
                                                float* __restrict__ out) {
  int wave = (blockIdx.x * blockDim.x + threadIdx.x) >> 5;  // 0..1023
  int lane = threadIdx.x & 31;
  int nt = wave >> 4;    // 0..63  (N tile)
  int ot = wave & 15;    // 0..15  (OUT tile)
  int row = lane & 15;
  int hi  = lane >> 4;

  const float* fp = F + (size_t)(nt * 16 + row) * PACK + 8 * hi;
  const float* cp = C + (size_t)(ot * 16 + row) * PACK + 8 * hi;

  float accf[8] = {0.f, 0.f, 0.f, 0.f, 0.f, 0.f, 0.f, 0.f};

#pragma unroll 4
  for (int i = 0; i < IN_DIM; ++i) {
    const float* fpi = fp + (size_t)i * (N_ROWS * PACK);
    const float* cpi = cp + (size_t)i * (OUT_DIM * PACK);

    // stream-ahead into WGP$/L2 (F lines are shared by all 8 waves per block)
    __builtin_prefetch(fpi + 8 * (N_ROWS * PACK), 0, 3);
    __builtin_prefetch(cpi + 8 * (OUT_DIM * PACK), 0, 3);

    v4f fa  = *(const v4f*)(fpi);        // (k0,k1,k4,k5) for this lane-half
    v2f fa2 = *(const v2f*)(fpi + 4);    // (k8,k9)
    v4f ca  = *(const v4f*)(cpi);
    v2f ca2 = *(const v2f*)(cpi + 4);

    v2f a0 = __builtin_shufflevector(fa, fa, 0, 1);
    v2f a1 = __builtin_shufflevector(fa, fa, 2, 3);
    v2f b0 = __builtin_shufflevector(ca, ca, 0, 1);
    v2f b1 = __builtin_shufflevector(ca, ca, 2, 3);

    v8f c = {};
    c = __builtin_amdgcn_wmma_f32_16x16x4_f32(false, a0, false, b0,
                                              (short)0, c, false, false);
    c = __builtin_amdgcn_wmma_f32_16x16x4_f32(false, a1, false, b1,
                                              (short)0, c, false, false);
    c = __builtin_amdgcn_wmma_f32_16x16x4_f32(false, fa2, false, ca2,
                                              (short)0, c, false, false);

    // coeffs pre-scaled by 64: c[] == act*64; floor is integer-valued float
#pragma unroll
    for (int r = 0; r < 8; ++r)
      accf[r] += floorf(c[r]);
  }

#pragma unroll
  for (int r = 0; r < 8; ++r) {
    int s = (int)accf[r];                          // exact (integer-valued)
    int w = ((s + 32768) & 0xFFFF) - 32768;        // final quant wrap (exact)
    int n = nt * 16 + r + 8 * hi;
    int o = ot * 16 + (lane & 15);
    out[n * OUT_DIM + o] = (float)w * (1.0f / 64.0f);
  }
}

// ---------------------------------------------------------------------------
extern "C" void kernel_launch(void* const* d_in, const int* in_sizes, int n_in,
                              void* d_out, int out_size, void* d_ws, size_t ws_size,
                              hipStream_t stream) {
  (void)in_sizes; (void)n_in; (void)out_size; (void)ws_size;

  const float* x    = (const float*)d_in[0];  // (N, IN)
  const float* bw   = (const float*)d_in[1];  // (OUT, IN)
  const float* sw   = (const float*)d_in[2];  // (OUT, IN, 8)
  const float* sc   = (const float*)d_in[3];  // (OUT, IN)
  const float* sel  = (const float*)d_in[4];  // (OUT, IN)
  const float* grid = (const float*)d_in[5];  // (IN, 12)

  float* F = (float*)d_ws;                               // 16.8 MB
  float* C = F + (size_t)IN_DIM * N_ROWS * PACK;         // +4.2 MB
  float* R = C + (size_t)IN_DIM * OUT_DIM * PACK;        // +32 KB

  kan_recips<<<1, IN_DIM, 0, stream>>>(grid, R);
  kan_features<<<(N_ROWS * IN_DIM) / 256, 256, 0, stream>>>(x, grid, R, F);
  kan_coefs<<<(OUT_DIM * IN_DIM) / 256, 256, 0, stream>>>(bw, sw, sc, sel, C);

  // 1024 tile-waves, 8 waves (256 threads) per block
  kan_wmma<<<(N_ROWS / 16) * (OUT_DIM / 16) / 8, 256, 0, stream>>>(F, C, (float*)d_out);
}